// CrossAttentionGNNConv_463856468619
// MI455X (gfx1250) — compile-verified
//
#include <hip/hip_runtime.h>
#include <hip/hip_bf16.h>
#include <math.h>

// ---------------------------------------------------------------------------
// CDNA5 (gfx1250) implementation of CrossAttentionGNNConv
//   - 6 node projections via V_WMMA_F32_16X16X4_F32 (full fp32 accumulate)
//   - per-edge dot-product scores, segment softmax via sign-split atomic max
//   - fused normalize + message scatter with global_atomic_add_f32
// ---------------------------------------------------------------------------

typedef __attribute__((ext_vector_type(2))) float v2f;
typedef __attribute__((ext_vector_type(8))) float v8f;

#define D 64

// ---------------------------------------------------------------------------
// GEMM: out[n x 64] = X[n x 64] @ W^T + bias     (W is [64 x 64], row-major)
// One wave32 computes a 16x16 output tile; 4 waves/block cover the 64 cols of
// one 16-row stripe.  K is stepped by 4 -> 16 v_wmma_f32_16x16x4_f32 per wave.
//
// A-matrix (16x4 f32) lane layout (ISA 7.12.2):
//   lanes 0-15 : v0 = A[m][k+0], v1 = A[m][k+1]
//   lanes16-31 : v0 = A[m][k+2], v1 = A[m][k+3]        (m = lane & 15)
// B-matrix (4x16) mirrors it with B[k][nc] = W[nc][k]  (nc = wave*16 + lane&15)
// C/D (16x16 f32): VGPR j -> row (j + 8*(lane>>4)), col = lane & 15.
// ---------------------------------------------------------------------------
__global__ __launch_bounds__(128)
void gemm_nt_64_wmma(const float* __restrict__ X, const float* __restrict__ W,
                     const float* __restrict__ bias, float* __restrict__ out,
                     int n) {
  const int lane = threadIdx.x & 31;
  const int wave = threadIdx.x >> 5;      // column tile 0..3
  const int half = lane >> 4;             // 0 / 1
  const int l16  = lane & 15;

  const int m  = blockIdx.x * 16 + l16;            // output row this lane owns in A
  const int ml = m < n ? m : (n - 1);              // clamped for safe loads
  const int nc = wave * 16 + l16;                  // output column this lane owns in B

  const float* __restrict__ xrow = X + (size_t)ml * D;
  const float* __restrict__ wrow = W + (size_t)nc * D;

  v8f c = {};
#pragma unroll
  for (int k = 0; k < D; k += 4) {
    const int kk = k + 2 * half;                   // even -> 8B aligned float2
    const float2 av = *reinterpret_cast<const float2*>(xrow + kk);
    const float2 bv = *reinterpret_cast<const float2*>(wrow + kk);
    v2f a; a.x = av.x; a.y = av.y;
    v2f b; b.x = bv.x; b.y = bv.y;
    c = __builtin_amdgcn_wmma_f32_16x16x4_f32(false, a, false, b,
                                              (short)0, c, false, false);
  }

  const float bb = bias ? bias[nc] : 0.0f;
  const int rowBase = blockIdx.x * 16;
#pragma unroll
  for (int j = 0; j < 8; ++j) {
    const int r = rowBase + j + 8 * half;          // D-matrix row for c[j]
    if (r < n) out[(size_t)r * D + nc] = c[j] + bb;
  }
}

// ---------------------------------------------------------------------------
// Init kernels
// ---------------------------------------------------------------------------
__global__ void init_stats(float* mA, float* mB, float* dA, float* dB, int n) {
  int i = blockIdx.x * blockDim.x + threadIdx.x;
  if (i >= n) return;
  mA[i] = -INFINITY; mB[i] = -INFINITY;
  dA[i] = 0.0f;      dB[i] = 0.0f;
}

__global__ void zero_f32(float* p, int total) {
  int i = blockIdx.x * blockDim.x + threadIdx.x;
  if (i < total) p[i] = 0.0f;
}

// ---------------------------------------------------------------------------
// Sign-split float atomic max (native int/uint atomics; correct for mixed
// signs with init = -inf, the classic bit-pattern monotonicity trick).
// ---------------------------------------------------------------------------
__device__ __forceinline__ void atomicMaxF(float* addr, float v) {
  if (v >= 0.0f) atomicMax(reinterpret_cast<int*>(addr), __float_as_int(v));
  else           atomicMin(reinterpret_cast<unsigned int*>(addr), __float_as_uint(v));
}

// ---------------------------------------------------------------------------
// Edge scores: sA[e] = <qa[row], ka[col]> * scale, sB likewise;
// running segment max via atomics.  One thread per edge, float4 gathers
// (all projected tables fit in the 192MB L2, so gathers are L2-resident).
// ---------------------------------------------------------------------------
__global__ __launch_bounds__(256)
void edge_scores(const int* __restrict__ ei,
                 const float* __restrict__ qa, const float* __restrict__ ka,
                 const float* __restrict__ qb, const float* __restrict__ kb,
                 float* __restrict__ sA, float* __restrict__ sB,
                 float* __restrict__ mA, float* __restrict__ mB, int E) {
  int e = blockIdx.x * blockDim.x + threadIdx.x;
  if (e >= E) return;
  const int r = ei[e];
  const int c = ei[E + e];

  const float4* __restrict__ qav = reinterpret_cast<const float4*>(qa + (size_t)r * D);
  const float4* __restrict__ kav = reinterpret_cast<const float4*>(ka + (size_t)c * D);
  const float4* __restrict__ qbv = reinterpret_cast<const float4*>(qb + (size_t)r * D);
  const float4* __restrict__ kbv = reinterpret_cast<const float4*>(kb + (size_t)c * D);

  float a = 0.0f, b = 0.0f;
#pragma unroll
  for (int i = 0; i < D / 4; ++i) {
    const float4 x = qav[i], y = kav[i];
    a = fmaf(x.x, y.x, a); a = fmaf(x.y, y.y, a);
    a = fmaf(x.z, y.z, a); a = fmaf(x.w, y.w, a);
    const float4 u = qbv[i], v = kbv[i];
    b = fmaf(u.x, v.x, b); b = fmaf(u.y, v.y, b);
    b = fmaf(u.z, v.z, b); b = fmaf(u.w, v.w, b);
  }
  const float scale = 0.125f;  // 1/sqrt(64)
  a *= scale; b *= scale;
  sA[e] = a; sB[e] = b;
  atomicMaxF(mA + r, a);
  atomicMaxF(mB + r, b);
}

// ---------------------------------------------------------------------------
// exp(s - max) in place + denominator accumulation
// ---------------------------------------------------------------------------
__global__ __launch_bounds__(256)
void edge_exp(const int* __restrict__ ei,
              float* __restrict__ sA, float* __restrict__ sB,
              const float* __restrict__ mA, const float* __restrict__ mB,
              float* __restrict__ dA, float* __restrict__ dB, int E) {
  int e = blockIdx.x * blockDim.x + threadIdx.x;
  if (e >= E) return;
  const int r = ei[e];
  const float a = __expf(sA[e] - mA[r]);
  const float b = __expf(sB[e] - mB[r]);
  sA[e] = a; sB[e] = b;
  atomicAdd(dA + r, a);
  atomicAdd(dB + r, b);
}

// ---------------------------------------------------------------------------
// Fused normalize + message scatter.  64 lanes per edge (feature j per lane):
// coalesced gather of the transformed source rows, two no-return
// global_atomic_add_f32 per lane.
// ---------------------------------------------------------------------------
__global__ __launch_bounds__(256)
void edge_scatter(const int* __restrict__ ei,
                  const float* __restrict__ sA, const float* __restrict__ sB,
                  const float* __restrict__ dA, const float* __restrict__ dB,
                  const float* __restrict__ wt, const float* __restrict__ wx,
                  float* __restrict__ out_x, float* __restrict__ out_t, int E) {
  const int t = blockIdx.x * blockDim.x + threadIdx.x;
  const int e = t >> 6;          // edge id
  const int j = t & (D - 1);     // feature id
  if (e >= E) return;
  const int r = ei[e];
  const int c = ei[E + e];
  const float wAv = sA[e] / dA[r];
  const float wBv = sB[e] / dB[r];
  atomicAdd(out_t + (size_t)r * D + j, wAv * wt[(size_t)c * D + j]);
  atomicAdd(out_x + (size_t)r * D + j, wBv * wx[(size_t)c * D + j]);
}

// ---------------------------------------------------------------------------
// Host-side launcher
// ---------------------------------------------------------------------------
extern "C" void kernel_launch(void* const* d_in, const int* in_sizes, int n_in,
                              void* d_out, int out_size, void* d_ws, size_t ws_size,
                              hipStream_t stream) {
  const float* x_src = (const float*)d_in[0];
  const float* x_tgt = (const float*)d_in[1];
  const float* t_src = (const float*)d_in[2];
  const float* t_tgt = (const float*)d_in[3];
  const int*   ei    = (const int*)  d_in[4];   // [2, E] int32
  const float* W_x   = (const float*)d_in[5];
  const float* W_t   = (const float*)d_in[6];
  const float* Ka_W  = (const float*)d_in[7];
  const float* Ka_b  = (const float*)d_in[8];
  const float* Qa_W  = (const float*)d_in[9];
  const float* Qa_b  = (const float*)d_in[10];
  const float* Kb_W  = (const float*)d_in[11];
  const float* Kb_b  = (const float*)d_in[12];
  const float* Qb_W  = (const float*)d_in[13];
  const float* Qb_b  = (const float*)d_in[14];

  const int n = in_sizes[0] / D;     // 50000
  const int E = in_sizes[4] / 2;     // 800000

  // workspace carve-out (floats)
  float* ws = (float*)d_ws;
  float* qa = ws;                    ws += (size_t)n * D;
  float* ka = ws;                    ws += (size_t)n * D;
  float* qb = ws;                    ws += (size_t)n * D;
  float* kb = ws;                    ws += (size_t)n * D;
  float* wt = ws;                    ws += (size_t)n * D;
  float* wx = ws;                    ws += (size_t)n * D;
  float* sA = ws;                    ws += E;
  float* sB = ws;                    ws += E;
  float* mA = ws;                    ws += n;
  float* mB = ws;                    ws += n;
  float* dA = ws;                    ws += n;
  float* dB = ws;                    ws += n;

  float* out_x = (float*)d_out;              // first  n*D floats
  float* out_t = (float*)d_out + (size_t)n * D;  // second n*D floats

  // ---- node projections (WMMA f32) ----
  const int gemmGrid = (n + 15) / 16;        // 3125 for n=50000 (exact)
  gemm_nt_64_wmma<<<gemmGrid, 128, 0, stream>>>(t_tgt, Qa_W, Qa_b, qa, n);
  gemm_nt_64_wmma<<<gemmGrid, 128, 0, stream>>>(t_src, Ka_W, Ka_b, ka, n);
  gemm_nt_64_wmma<<<gemmGrid, 128, 0, stream>>>(x_tgt, Qb_W, Qb_b, qb, n);
  gemm_nt_64_wmma<<<gemmGrid, 128, 0, stream>>>(x_src, Kb_W, Kb_b, kb, n);
  gemm_nt_64_wmma<<<gemmGrid, 128, 0, stream>>>(t_src, W_t,  nullptr, wt, n);
  gemm_nt_64_wmma<<<gemmGrid, 128, 0, stream>>>(x_src, W_x,  nullptr, wx, n);

  // ---- init outputs / softmax stats ----
  const int outTotal = 2 * n * D;
  zero_f32  <<<(outTotal + 255) / 256, 256, 0, stream>>>((float*)d_out, outTotal);
  init_stats<<<(n + 255) / 256, 256, 0, stream>>>(mA, mB, dA, dB, n);

  // ---- segment softmax over edges ----
  edge_scores<<<(E + 255) / 256, 256, 0, stream>>>(ei, qa, ka, qb, kb,
                                                   sA, sB, mA, mB, E);
  edge_exp   <<<(E + 255) / 256, 256, 0, stream>>>(ei, sA, sB, mA, mB, dA, dB, E);

  // ---- normalize + scatter messages ----
  const long long threadsScatter = (long long)E * D;
  edge_scatter<<<(int)((threadsScatter + 255) / 256), 256, 0, stream>>>(
      ei, sA, sB, dA, dB, wt, wx, out_x, out_t, E);
}